// NMS_45028437131298
// MI455X (gfx1250) — compile-verified
//
#include <hip/hip_runtime.h>
#include <stdint.h>

// ---------------- problem constants (match reference) ----------------
#define NBOX   25200
#define NCH    85
#define NCLS   80
#define KTOP   1024      // padded TOPK (real TOPK = 1000)
#define KREAL  1000
#define MAXDET 300
#define CONF_T 0.25f
#define IOU_T  0.45f
#define MAXWH  4096.0f
#define K1_BOXES 128

typedef float v2f __attribute__((ext_vector_type(2)));
typedef float v8f __attribute__((ext_vector_type(8)));

__device__ __forceinline__ uint32_t orderKey(float f) {
  uint32_t u = __float_as_uint(f);
  return (u & 0x80000000u) ? ~u : (u | 0x80000000u);
}

// =====================================================================
// Kernel 1: decode. Stage 128 boxes (43.5KB) into LDS via CDNA5 async
// global->LDS copies (non-temporal: pred is streamed exactly once),
// then compute xyxy box, best-class score, class id.
// =====================================================================
__global__ void __launch_bounds__(K1_BOXES)
decode_kernel(const float* __restrict__ pred,
              float4* __restrict__ boxAll,
              float* __restrict__ scoreAll,
              int* __restrict__ clsAll,
              int blocksPerImg) {
  __shared__ float lds[K1_BOXES * NCH];

  int blk = blockIdx.x;
  int img = blk / blocksPerImg;
  int cb  = blk % blocksPerImg;
  int b0  = cb * K1_BOXES;
  int nb  = min(K1_BOXES, NBOX - b0);
  int tid = threadIdx.x;

  const float* src = pred + ((size_t)img * NBOX + b0) * NCH;
  int totalBytes = nb * NCH * 4;              // multiple of 16 for nb%4==0
  uint32_t ldsBase = (uint32_t)(uintptr_t)(&lds[0]);
  uint64_t sbase   = (uint64_t)(uintptr_t)src;

  for (int off = tid * 16; off < totalBytes; off += K1_BOXES * 16) {
    uint32_t la = ldsBase + (uint32_t)off;
    uint32_t vo = (uint32_t)off;
    asm volatile("global_load_async_to_lds_b128 %0, %1, %2 offset:0 th:TH_LOAD_NT"
                 :: "v"(la), "v"(vo), "s"(sbase) : "memory");
  }
  asm volatile("s_wait_asynccnt 0x0" ::: "memory");
  __syncthreads();

  if (tid < nb) {
    const float* s = &lds[tid * NCH];
    float cx = s[0], cy = s[1], w = s[2], h = s[3], obj = s[4];
    float mv = s[5]; int mi = 0;
    #pragma unroll 8
    for (int c = 1; c < NCLS; ++c) {
      float v = s[5 + c];
      if (v > mv) { mv = v; mi = c; }          // strict > : first argmax
    }
    float sc = obj * mv;
    float st = (sc > CONF_T) ? sc : -1.0f;
    size_t g = (size_t)img * NBOX + b0 + tid;
    boxAll[g]  = make_float4(cx - 0.5f * w, cy - 0.5f * h,
                             cx + 0.5f * w, cy + 0.5f * h);
    scoreAll[g] = st;
    clsAll[g]   = mi;
  }
}

// =====================================================================
// Kernel 2: per-image exact top-1000 by 4-pass radix select + bitonic
// sort (score desc, idx asc). One 1024-thread block per image.
// =====================================================================
__global__ void __launch_bounds__(1024)
topk_kernel(const float4* __restrict__ boxAll,
            const float* __restrict__ scoreAll,
            const int* __restrict__ clsAll,
            float* __restrict__ valsK, float* __restrict__ clsK,
            float4* __restrict__ boxK, float4* __restrict__ bbK) {
  __shared__ uint32_t hist[256];
  __shared__ float svals[KTOP];
  __shared__ int   sidx[KTOP];
  __shared__ uint32_t shPrefix;
  __shared__ int shWant, cntSel, cntEq;

  int img = blockIdx.x;
  int tid = threadIdx.x;
  const float* sc = scoreAll + (size_t)img * NBOX;

  if (tid == 0) { shPrefix = 0u; shWant = KREAL; cntSel = 0; cntEq = 0; }
  __syncthreads();

  // ---- radix select: find key of the 1000th largest ----
  for (int shift = 24; shift >= 0; shift -= 8) {
    if (tid < 256) hist[tid] = 0;
    __syncthreads();
    uint32_t pfx   = shPrefix;
    uint32_t pmask = (shift == 24) ? 0u : (0xFFFFFFFFu << (shift + 8));
    for (int i = tid; i < NBOX; i += 1024) {
      uint32_t k = orderKey(sc[i]);
      if ((k & pmask) == pfx)
        atomicAdd(&hist[(k >> shift) & 255u], 1u);
    }
    __syncthreads();
    if (tid == 0) {
      int want = shWant;
      uint32_t running = 0; int digit = 0;
      for (int b = 255; b >= 0; --b) {
        uint32_t h = hist[b];
        if (running + h >= (uint32_t)want) { digit = b; break; }
        running += h;
      }
      shWant   = want - (int)running;
      shPrefix = shPrefix | ((uint32_t)digit << shift);
    }
    __syncthreads();
  }
  uint32_t T = shPrefix;
  int want   = shWant;

  // ---- compact the 1000 selected (val, idx) pairs into LDS ----
  for (int i = tid; i < KTOP; i += 1024) { svals[i] = -3.0e38f; sidx[i] = 0x7FFFFFFF; }
  __syncthreads();
  for (int i = tid; i < NBOX; i += 1024) {
    float v = sc[i];
    uint32_t k = orderKey(v);
    if (k > T) {
      int p = atomicAdd(&cntSel, 1);
      if (p < KTOP) { svals[p] = v; sidx[p] = i; }
    } else if (k == T) {
      int e = atomicAdd(&cntEq, 1);
      if (e < want) {
        int p = atomicAdd(&cntSel, 1);
        if (p < KTOP) { svals[p] = v; sidx[p] = i; }
      }
    }
  }
  __syncthreads();

  // ---- bitonic sort 1024: score desc, idx asc ----
  for (int k = 2; k <= KTOP; k <<= 1) {
    for (int j = k >> 1; j > 0; j >>= 1) {
      __syncthreads();
      int ixj = tid ^ j;
      if (ixj > tid) {
        float v0 = svals[tid], v1 = svals[ixj];
        int   i0 = sidx[tid],  i1 = sidx[ixj];
        bool g01 = (v0 > v1) || (v0 == v1 && i0 < i1);  // tid earlier
        bool g10 = (v1 > v0) || (v1 == v0 && i1 < i0);  // ixj earlier
        bool ascend = ((tid & k) == 0);
        if (ascend ? g10 : g01) {
          svals[tid] = v1; svals[ixj] = v0;
          sidx[tid]  = i1; sidx[ixj]  = i0;
        }
      }
    }
  }
  __syncthreads();

  // ---- gather boxes/classes for the K candidates ----
  {
    float v; int id;
    if (tid < KREAL) { v = svals[tid]; id = sidx[tid]; }
    else             { v = -1.0f;      id = -1; }
    float4 bx = make_float4(0.f, 0.f, 0.f, 0.f);
    float cf = 0.0f;
    if (id >= 0 && id < NBOX) {
      bx = boxAll[(size_t)img * NBOX + id];
      cf = (float)clsAll[(size_t)img * NBOX + id];
    }
    size_t o = (size_t)img * KTOP + tid;
    valsK[o] = v; clsK[o] = cf; boxK[o] = bx;
    float offc = cf * MAXWH;                   // class-aware offset trick
    bbK[o] = make_float4(bx.x + offc, bx.y + offc, bx.z + offc, bx.w + offc);
  }
}

// =====================================================================
// Kernel 3: suppression bitmask. One wave per 16x16 IoU tile.
// WMMA f32 16x16x4 computes the outer sum area_i + area_j; the C-matrix
// lane layout is reused directly for the per-pair intersection math and
// ballot-based bit packing. The IoU>thres test is divide-free:
//   inter/(union+eps) > t  <=>  inter > t*(union+eps)   (denominator > 0)
// mask[i][w] (w in 0..63) holds bits for j in [16w,16w+16);
// bit set => i suppresses j (only j > i set).
// =====================================================================
__global__ void __launch_bounds__(256)
mask_kernel(const float4* __restrict__ bbK, uint32_t* __restrict__ mask) {
  int wave = (blockIdx.x * blockDim.x + threadIdx.x) >> 5;
  int lane = threadIdx.x & 31;
  int img  = wave >> 12;            // 64*64 = 4096 tiles per image
  int rem  = wave & 4095;
  int ti   = rem >> 6;
  int tj   = rem & 63;
  int lane16 = lane & 15;
  int hi     = lane >> 4;

  const float4* bb = bbK + (size_t)img * KTOP;
  float4 jb = bb[tj * 16 + lane16];
  float areaJ = (jb.z - jb.x) * (jb.w - jb.y);
  float4 ab = bb[ti * 16 + lane16];
  float areaA = (ab.z - ab.x) * (ab.w - ab.y);

  // A (16x4): lanes0-15 hold K=0,1 ; lanes16-31 hold K=2,3 (zeros)
  // B (4x16): row0 = ones, row1 = areaJ, rows 2,3 = zeros
  v2f A, Bv;
  A.x  = hi ? 0.0f : areaA;
  A.y  = hi ? 0.0f : 1.0f;
  Bv.x = hi ? 0.0f : 1.0f;
  Bv.y = hi ? 0.0f : areaJ;

  v8f acc = {0.f, 0.f, 0.f, 0.f, 0.f, 0.f, 0.f, 0.f};
#if __has_builtin(__builtin_amdgcn_wmma_f32_16x16x4_f32)
  acc = __builtin_amdgcn_wmma_f32_16x16x4_f32(false, A, false, Bv,
                                              (short)0, acc, false, false);
#else
  // VALU fallback: acc[v] = area(row v + 8*hi) + areaJ
  #pragma unroll
  for (int v = 0; v < 8; ++v) {
    float4 rb = bb[ti * 16 + v + 8 * hi];
    acc[v] = (rb.z - rb.x) * (rb.w - rb.y) + areaJ;
  }
#endif

  float4 ib[8];
  #pragma unroll
  for (int v = 0; v < 8; ++v) ib[v] = bb[ti * 16 + v + 8 * hi];

  int gj = tj * 16 + lane16;
  uint32_t* mrow = mask + (size_t)img * KTOP * 64;
  #pragma unroll
  for (int v = 0; v < 8; ++v) {
    int gi = ti * 16 + v + 8 * hi;
    float lx = fmaxf(ib[v].x, jb.x);
    float ly = fmaxf(ib[v].y, jb.y);
    float rx = fminf(ib[v].z, jb.z);
    float ry = fminf(ib[v].w, jb.w);
    float iw = fmaxf(rx - lx, 0.0f);
    float ihh = fmaxf(ry - ly, 0.0f);
    float inter = iw * ihh;
    // divide-free IoU threshold test (denominator strictly positive)
    bool cond = (inter > IOU_T * (acc[v] - inter + 1e-9f)) && (gj > gi);
    uint32_t bal = __builtin_amdgcn_ballot_w32(cond);
    // low 16 bits: lanes 0-15 -> row ti*16+v ; high: row ti*16+v+8
    if (lane == 2 * v)     mrow[(size_t)(ti * 16 + v    ) * 64 + tj] = bal & 0xFFFFu;
    if (lane == 2 * v + 1) mrow[(size_t)(ti * 16 + v + 8) * 64 + tj] = bal >> 16;
  }
}

// =====================================================================
// Kernel 4: greedy scan (one wave per image, remv bitmask in 2 VGPRs
// per lane), then ballot-based prefix-sum compaction and output write.
// =====================================================================
__global__ void __launch_bounds__(1024)
scan_out_kernel(const float* __restrict__ valsK, const float* __restrict__ clsK,
                const float4* __restrict__ boxK, const uint32_t* __restrict__ mask,
                float* __restrict__ out) {
  __shared__ float vls[KTOP];
  __shared__ uint8_t keep[KTOP];
  __shared__ int waveSum[32];

  int img  = blockIdx.x;
  int tid  = threadIdx.x;
  int lane = tid & 31;
  int wv   = tid >> 5;

  for (int i = tid; i < KTOP; i += 1024) vls[i] = valsK[(size_t)img * KTOP + i];
  __syncthreads();

  if (tid < 32) {
    const uint32_t* m = mask + (size_t)img * KTOP * 64;
    uint32_t r0 = 0, r1 = 0;               // lane owns words 2*lane, 2*lane+1
    for (int i = 0; i < KTOP; ++i) {
      int word  = i >> 4;
      int owner = word >> 1;
      int selhi = word & 1;
      uint32_t rr = selhi ? r1 : r0;
      int rbit = (int)((rr >> (i & 15)) & 1u);
      int rem  = __shfl(rbit, owner, 32);
      int kp   = (!rem) && (vls[i] > 0.0f);
      if (lane == 0) keep[i] = (uint8_t)kp;
      if (kp) {
        r0 |= m[(size_t)i * 64 + 2 * lane];
        r1 |= m[(size_t)i * 64 + 2 * lane + 1];
      }
    }
  }
  __syncthreads();

  // prefix sum of keep flags across 1024 threads
  int k = keep[tid] ? 1 : 0;
  uint32_t bal = __builtin_amdgcn_ballot_w32(k != 0);
  uint32_t below = (lane == 0) ? 0u : (0xFFFFFFFFu >> (32 - lane));
  int lanePfx = __popc(bal & below);
  if (lane == 31) waveSum[wv] = __popc(bal);
  __syncthreads();
  if (tid < 32) {
    int s = waveSum[tid];
    for (int off = 1; off < 32; off <<= 1) {
      int t = __shfl_up(s, off, 32);
      if (lane >= off) s += t;
    }
    waveSum[tid] = s;                      // inclusive scan
  }
  __syncthreads();
  int base = (wv == 0) ? 0 : waveSum[wv - 1];
  int pos  = base + lanePfx;

  float* o = out + (size_t)img * MAXDET * 6;
  for (int i = tid; i < MAXDET * 6; i += 1024) o[i] = 0.0f;
  __syncthreads();

  if (k && pos < MAXDET) {
    float4 bx = boxK[(size_t)img * KTOP + tid];
    float  cf = clsK[(size_t)img * KTOP + tid];
    float  v  = vls[tid];
    float* row = o + pos * 6;
    row[0] = bx.x; row[1] = bx.y; row[2] = bx.z; row[3] = bx.w;
    row[4] = v;    row[5] = cf;
  }
}

// =====================================================================
extern "C" void kernel_launch(void* const* d_in, const int* in_sizes, int n_in,
                              void* d_out, int out_size, void* d_ws, size_t ws_size,
                              hipStream_t stream) {
  (void)n_in; (void)out_size; (void)ws_size;
  const float* pred = (const float*)d_in[0];
  int B = in_sizes[0] / (NBOX * NCH);      // 16 for reference shapes

  size_t nAll = (size_t)B * NBOX;
  size_t off = 0;
  char* ws = (char*)d_ws;
  auto take = [&](size_t bytes) -> char* {
    char* p = ws + off;
    off += (bytes + 255) & ~(size_t)255;
    return p;
  };
  float4*   boxAll   = (float4*)take(nAll * sizeof(float4));
  float*    scoreAll = (float*) take(nAll * sizeof(float));
  int*      clsAll   = (int*)   take(nAll * sizeof(int));
  float*    valsK    = (float*) take((size_t)B * KTOP * sizeof(float));
  float*    clsK     = (float*) take((size_t)B * KTOP * sizeof(float));
  float4*   boxK     = (float4*)take((size_t)B * KTOP * sizeof(float4));
  float4*   bbK      = (float4*)take((size_t)B * KTOP * sizeof(float4));
  uint32_t* maskP    = (uint32_t*)take((size_t)B * KTOP * 64 * sizeof(uint32_t));

  int blocksPerImg = (NBOX + K1_BOXES - 1) / K1_BOXES;   // 197
  decode_kernel<<<B * blocksPerImg, K1_BOXES, 0, stream>>>(
      pred, boxAll, scoreAll, clsAll, blocksPerImg);

  topk_kernel<<<B, 1024, 0, stream>>>(
      boxAll, scoreAll, clsAll, valsK, clsK, boxK, bbK);

  int nWaves = B * 64 * 64;                               // 16x16 tiles
  mask_kernel<<<nWaves / 8, 256, 0, stream>>>(bbK, maskP);

  scan_out_kernel<<<B, 1024, 0, stream>>>(
      valsK, clsK, boxK, maskP, (float*)d_out);
}